// OurModel_8160437862737
// MI455X (gfx1250) — compile-verified
//
#include <hip/hip_runtime.h>
#include <math.h>

typedef __attribute__((ext_vector_type(2))) float v2f;
typedef __attribute__((ext_vector_type(8))) float v8f;

#define KNN   20           // neighbors used for features
#define KNNP  21           // list size: top-20 + the self point
#define EMB   16
#define NWAVES 8

// Fully-unrolled sorted insert (ascending key). Arrays stay in VGPRs because
// every index is a compile-time constant after unrolling.
__device__ __forceinline__ void topk_insert(float (&td)[KNNP], int (&ti)[KNNP],
                                            float d2, int m) {
    if (d2 >= td[KNNP - 1]) return;
#pragma unroll
    for (int j = KNNP - 1; j >= 1; --j) {
        float pd = td[j - 1];
        int   pi = ti[j - 1];
        bool keep     = (td[j] <= d2);   // slot stays (among the k smaller ones)
        bool fromprev = (pd > d2);       // shifted down from j-1
        td[j] = keep ? td[j] : (fromprev ? pd : d2);
        ti[j] = keep ? ti[j] : (fromprev ? pi : m);
    }
    if (td[0] > d2) { td[0] = d2; ti[0] = m; }
}

// One 16-point x 16-query tile: 5x V_WMMA_F32_16X16X4_F32.
//   1 norm-broadcast WMMA  (A = per-half partial |e|^2, B = ones in k=0,2 rows)
//   4 dot-chunk WMMAs with B pre-scaled by -2
// Accumulator holds key = |e|^2 - 2<e,q>  (== d^2 - |q|^2, same ordering).
__device__ __forceinline__ v8f tile_keys(const float* __restrict__ prow,
                                         const v2f bqm[4], v2f bone) {
    v2f a0 = *(const v2f*)(prow + 0);
    v2f a1 = *(const v2f*)(prow + 4);
    v2f a2 = *(const v2f*)(prow + 8);
    v2f a3 = *(const v2f*)(prow + 12);

    // Half-norm of this lane's point row (packed FMA chain).
    v2f s = a0 * a0;
    s = a1 * a1 + s;
    s = a2 * a2 + s;
    s = a3 * a3 + s;
    v2f an; an.x = s.x + s.y; an.y = 0.0f;  // A[m][k=0 or 2] = half norm

    v8f c = {};
    c = __builtin_amdgcn_wmma_f32_16x16x4_f32(false, an, false, bone,  (short)0, c, false, false);
    c = __builtin_amdgcn_wmma_f32_16x16x4_f32(false, a0, false, bqm[0], (short)0, c, false, false);
    c = __builtin_amdgcn_wmma_f32_16x16x4_f32(false, a1, false, bqm[1], (short)0, c, false, false);
    c = __builtin_amdgcn_wmma_f32_16x16x4_f32(false, a2, false, bqm[2], (short)0, c, false, false);
    c = __builtin_amdgcn_wmma_f32_16x16x4_f32(false, a3, false, bqm[3], (short)0, c, false, false);
    return c;
}

// One block = one (16-query tile, side). 8 waves scan disjoint point tiles,
// two tiles per iteration so the two WMMA accumulator chains interleave.
__global__ __launch_bounds__(256) void knn_feats_kernel(
    const float* __restrict__ emb0, const float* __restrict__ emb1,
    const float* __restrict__ rctx0, const float* __restrict__ rctx1,
    const int* __restrict__ idx0, const int* __restrict__ idx1,
    float* __restrict__ feats, int N) {

    __shared__ float s_d2[16][16 * KNNP];
    __shared__ int   s_id[16][16 * KNNP];

    const int tid  = threadIdx.x;
    const int lane = tid & 31;
    const int wave = tid >> 5;
    const int lo16 = lane & 15;
    const int hi   = lane >> 4;            // 0 or 1 (lane half)
    const int side = blockIdx.x & 1;
    const int tile = blockIdx.x >> 1;

    const float* __restrict__ emb  = side ? emb1 : emb0;
    const float* __restrict__ rctx = side ? rctx1 : rctx0;
    const int*   __restrict__ idx  = side ? idx1 : idx0;

    // This lane's query (column N = lo16 of the WMMA tile).
    const int q    = tile * 16 + lo16;
    const int idxq = idx[q];
    const float* qrow = emb + (size_t)idxq * EMB;

    // Query squared norm (per lane; only needed for the final sims).
    float sa = 0.f;
#pragma unroll
    for (int d = 0; d < EMB; ++d) { float v = qrow[d]; sa = fmaf(v, v, sa); }

    // B fragments pre-scaled by -2: vgpr r of chunk c holds
    //   -2 * Q[dim = 4c + r + 2*hi][query lo16].
    v2f bqm[4];
#pragma unroll
    for (int c = 0; c < 4; ++c) {
        v2f t = *(const v2f*)(qrow + 4 * c + 2 * hi);
        bqm[c].x = -2.0f * t.x;
        bqm[c].y = -2.0f * t.y;
    }
    // Ones in rows k=0 (lanes 0-15) and k=2 (lanes 16-31): vgpr0 = 1 everywhere.
    v2f bone; bone.x = 1.0f; bone.y = 0.0f;

    float td[KNNP]; int ti[KNNP];
#pragma unroll
    for (int j = 0; j < KNNP; ++j) { td[j] = 3.4e38f; ti[j] = 0; }

    const int    ntiles = N >> 4;          // N % 16 == 0 for this problem
    const size_t rowoff = (size_t)lo16 * EMB + 2 * hi;

    int p = wave;
    for (; p + NWAVES < ntiles; p += 2 * NWAVES) {
        const float* prow0 = emb + (size_t)p * 16 * EMB + rowoff;
        const float* prow1 = prow0 + (size_t)NWAVES * 16 * EMB;
        __builtin_prefetch(prow1 + (size_t)NWAVES * 16 * EMB, 0, 1);
        __builtin_prefetch(prow1 + (size_t)2 * NWAVES * 16 * EMB, 0, 1);

        v8f c0 = tile_keys(prow0, bqm, bone);
        v8f c1 = tile_keys(prow1, bqm, bone);

        // D layout: vgpr v, lane l -> point m = 16p + v + 8*hi, query lo16.
        const int mb0 = p * 16 + 8 * hi;
        const int mb1 = mb0 + NWAVES * 16;
#pragma unroll
        for (int v = 0; v < 8; ++v) topk_insert(td, ti, c0[v], mb0 + v);
#pragma unroll
        for (int v = 0; v < 8; ++v) topk_insert(td, ti, c1[v], mb1 + v);
    }
    if (p < ntiles) {                      // wave-uniform tail (all lanes active)
        const float* prow0 = emb + (size_t)p * 16 * EMB + rowoff;
        v8f c0 = tile_keys(prow0, bqm, bone);
        const int mb0 = p * 16 + 8 * hi;
#pragma unroll
        for (int v = 0; v < 8; ++v) topk_insert(td, ti, c0[v], mb0 + v);
    }

    // Merge the 16 per-lane lists (8 waves x 2 lane-halves) per query via LDS.
    const int slot = wave * 2 + hi;        // 0..15
#pragma unroll
    for (int j = 0; j < KNNP; ++j) {
        s_d2[lo16][slot * KNNP + j] = td[j];
        s_id[lo16][slot * KNNP + j] = ti[j];
    }
    __syncthreads();

    if (tid < 16) {
        const int qi = tid;                // == this thread's lo16, so sa/idxq match
        float ftd[KNNP]; int fti[KNNP];
#pragma unroll
        for (int j = 0; j < KNNP; ++j) { ftd[j] = 3.4e38f; fti[j] = 0; }
        for (int c = 0; c < 16 * KNNP; ++c)
            topk_insert(ftd, fti, s_d2[qi][c], s_id[qi][c]);

        const int qg = tile * 16 + qi;
        const float* rrow = rctx + (size_t)qg * N;

        // Top-21 contains the self point exactly once (its d^2 ~ 0 is the global
        // min); dropping it reproduces the reference's "top-(K+1), drop first".
        float f1 = 0.f, f2n = 0.f, ssum = 0.f;
        float selv[KNNP]; bool val[KNNP];
#pragma unroll
        for (int j = 0; j < KNNP; ++j) {
            bool valid = (fti[j] != idxq);
            float d2   = sa + ftd[j];                 // restore true d^2
            float sim  = sqrtf(fmaxf(d2, 0.f)) + 0.001f;
            float w    = expf(-sim);
            float sel  = rrow[fti[j]];
            selv[j] = sel; val[j] = valid;
            if (valid) { f1 += w; f2n = fmaf(w, sel, f2n); ssum += sel; }
        }
        float mu  = ssum * (1.0f / KNN);
        float var = 0.f;
#pragma unroll
        for (int j = 0; j < KNNP; ++j) {
            if (val[j]) { float d = selv[j] - mu; var = fmaf(d, d, var); }
        }
        float f3 = sqrtf(var * (1.0f / (KNN - 1)));
        float f2 = f2n / f1;

        // feats layout: [a1, b1, a2, b2, a3, b3] -> col = 2*feat + side
        feats[qg * 6 + 0 + side] = f1;
        feats[qg * 6 + 2 + side] = f2;
        feats[qg * 6 + 4 + side] = f3;
    }
}

__global__ __launch_bounds__(256) void mlp_kernel(
    const float* __restrict__ feats,
    const float* __restrict__ mean_in, const float* __restrict__ std_in,
    const float* __restrict__ W1, const float* __restrict__ b1,
    const float* __restrict__ Wm, const float* __restrict__ bm,
    const float* __restrict__ Ws, const float* __restrict__ bs,
    float* __restrict__ out, int B) {
    int q = blockIdx.x * blockDim.x + threadIdx.x;
    if (q >= B) return;
    float f[8];
#pragma unroll
    for (int i = 0; i < 6; ++i) f[i] = feats[q * 6 + i];
    f[6] = mean_in[q];
    f[7] = std_in[q];
    float mo = bm[0], so = bs[0];
#pragma unroll 4
    for (int j = 0; j < 64; ++j) {
        float h = b1[j];
#pragma unroll
        for (int i = 0; i < 8; ++i) h = fmaf(f[i], W1[i * 64 + j], h);
        h = fmaxf(h, 0.f);
        mo = fmaf(h, Wm[j], mo);
        so = fmaf(h, Ws[j], so);
    }
    out[q]     = mo;
    out[B + q] = so;
}

extern "C" void kernel_launch(void* const* d_in, const int* in_sizes, int n_in,
                              void* d_out, int out_size, void* d_ws, size_t ws_size,
                              hipStream_t stream) {
    const float* emb0    = (const float*)d_in[0];
    const float* emb1    = (const float*)d_in[1];
    const float* rctx0   = (const float*)d_in[2];
    const float* rctx1   = (const float*)d_in[3];
    const int*   idx0    = (const int*)d_in[4];
    const int*   idx1    = (const int*)d_in[5];
    const float* mean_in = (const float*)d_in[6];
    const float* std_in  = (const float*)d_in[7];
    const float* W1      = (const float*)d_in[8];
    const float* b1      = (const float*)d_in[9];
    const float* Wm      = (const float*)d_in[10];
    const float* bm      = (const float*)d_in[11];
    const float* Ws      = (const float*)d_in[12];
    const float* bs      = (const float*)d_in[13];
    float* out   = (float*)d_out;
    float* feats = (float*)d_ws;              // B*6 floats of scratch

    const int B = in_sizes[4];                // 512
    const int N = in_sizes[2] / B;            // 50000 (divisible by 16)
    const int qtiles = B / 16;                // 32

    knn_feats_kernel<<<qtiles * 2, 256, 0, stream>>>(emb0, emb1, rctx0, rctx1,
                                                     idx0, idx1, feats, N);
    mlp_kernel<<<(B + 255) / 256, 256, 0, stream>>>(feats, mean_in, std_in,
                                                    W1, b1, Wm, bm, Ws, bs, out, B);
}